// Net_35502199669162
// MI455X (gfx1250) — compile-verified
//
#include <hip/hip_runtime.h>
#include <math.h>

typedef __attribute__((ext_vector_type(2))) float v2f;
typedef __attribute__((ext_vector_type(8))) float v8f;

#define WMMA_F32(a, b, c) \
  __builtin_amdgcn_wmma_f32_16x16x4_f32(false, (a), false, (b), (short)0, (c), false, false)

// gfx1250 async global->LDS copy (ASYNCcnt-tracked), per CDNA5 ISA
__device__ __forceinline__ unsigned lds_off(const void* p) {
  // flat LDS address: low 32 bits are the LDS-relative byte address
  return (unsigned)(unsigned long long)p;
}
__device__ __forceinline__ void async_copy_b128(unsigned lds_dst, const float* g) {
  asm volatile("global_load_async_to_lds_b128 %0, %1, off"
               :: "v"(lds_dst), "v"(g) : "memory");
}
__device__ __forceinline__ void wait_async0() {
  asm volatile("s_wait_asynccnt 0x0" ::: "memory");
}

// ---- problem sizes ----
#define Bn   1024
#define Gn   8
#define HIDn 512
#define Hn   256
#define Cn   28
#define Nrows (Bn * Gn)          // 8192
#define Z1n  1024
#define Z2n  512

// combinations(range(8), 2), lexicographic
__constant__ int c_aidx[Cn] = {0,0,0,0,0,0,0,1,1,1,1,1,1,2,2,2,2,2,3,3,3,3,4,4,4,5,5,6};
__constant__ int c_bidx[Cn] = {1,2,3,4,5,6,7,2,3,4,5,6,7,3,4,5,6,7,4,5,6,7,5,6,7,6,7,7};

// ---- workspace layout (floats) ----
#define WS_MU1    0            // 512
#define WS_RSTD1  512          // 512
#define WS_MU2    1024         // 256
#define WS_RSTD2  1280         // 256
#define WS_MUZ    1536         // 2048
#define WS_RSTDZ  3584         // 2048
#define WS_BIL    5632         // 224*1024 = 229376
#define WS_SHARED 235520       // max(A_tanh 4194304, z1 8388608) = 8388608
#define WS_H      8624128      // 8192*256 = 2097152
#define WS_HBN    10721280     // 2097152
#define WS_WT     12818432     // 56*256*256 = 3670016
#define WS_Z1M    16488448     // 1024*8*512 = 4194304
#define WS_Z2     20682752     // 1024*8*256 = 2097152

// ============================================================
// column mean / rstd over R rows of an (R x Ccols) row-major matrix
__global__ __launch_bounds__(256) void col_stats_kernel(
    const float* __restrict__ X, int R, int Ccols,
    float* __restrict__ mean, float* __restrict__ rstd) {
  __shared__ float ssum[8][32];
  __shared__ float ssq[8][32];
  const int c = threadIdx.x & 31;
  const int rg = threadIdx.x >> 5;
  const int col = blockIdx.x * 32 + c;
  float s = 0.f, q = 0.f;
  for (int r = rg; r < R; r += 8) {
    float v = X[(long)r * Ccols + col];
    s += v; q += v * v;
  }
  ssum[rg][c] = s; ssq[rg][c] = q;
  __syncthreads();
  if (rg == 0) {
#pragma unroll
    for (int i = 1; i < 8; ++i) { s += ssum[i][c]; q += ssq[i][c]; }
    float mu = s / (float)R;
    float var = q / (float)R - mu * mu;
    mean[col] = mu;
    rstd[col] = rsqrtf(var + 1e-5f);
  }
}

// A_tanh[r,k] = tanh(g1[k]*(x[r,k]-mu1[k])*rstd1[k] + b1[k]), 4-wide
__global__ __launch_bounds__(256) void tanh_bn1_kernel(
    const float* __restrict__ x, const float* __restrict__ g1,
    const float* __restrict__ b1, const float* __restrict__ mu1,
    const float* __restrict__ rstd1, float* __restrict__ A) {
  int i4 = blockIdx.x * 256 + threadIdx.x;   // < (8192*512)/4
  int base = i4 * 4;
  int k = base & (HIDn - 1);
  float4 xv = *(const float4*)&x[base];
  float4 gv = *(const float4*)&g1[k];
  float4 bv = *(const float4*)&b1[k];
  float4 mv = *(const float4*)&mu1[k];
  float4 rv = *(const float4*)&rstd1[k];
  float4 o;
  o.x = tanhf(gv.x * (xv.x - mv.x) * rv.x + bv.x);
  o.y = tanhf(gv.y * (xv.y - mv.y) * rv.y + bv.y);
  o.z = tanhf(gv.z * (xv.z - mv.z) * rv.z + bv.z);
  o.w = tanhf(gv.w * (xv.w - mv.w) * rv.w + bv.w);
  *(float4*)&A[base] = o;
}

// w_t[(c*2+dir), i, j] = w[i, j, dir, c]   (w is (H,H,2,C))
__global__ __launch_bounds__(256) void repack_w_kernel(
    const float* __restrict__ w, float* __restrict__ wt) {
  int idx = blockIdx.x * 256 + threadIdx.x;   // < 56*256*256
  int cd = idx >> 16;
  int rem = idx & 0xFFFF;
  int i = rem >> 8, j = rem & 255;
  int c = cd >> 1, dir = cd & 1;
  wt[idx] = w[((i * Hn + j) * 2 + dir) * Cn + c];
}

// ============================================================
// generic fp32 WMMA GEMM with double-buffered async global->LDS staging.
// C[g] = A[g] (MxK) * B[g] (KxN) + bias[g]
// block tile 128x64, 8 waves; each wave: 16 rows x 64 cols (4 wmma tiles)
__global__ __launch_bounds__(256) void wmma_gemm_f32(
    const float* __restrict__ A, int lda, int aG,
    const float* __restrict__ Bm, int ldb, int bG,
    float* __restrict__ Cm, int ldc, int cG,
    const float* __restrict__ bias, int biasG, int K) {
  __shared__ float As[2][128][20];   // row stride 80B (16B aligned, conflict-free)
  __shared__ float Bs[2][16][68];    // row stride 272B
  const int tid = threadIdx.x;
  const int lane = tid & 31;
  const int wave = tid >> 5;
  const int hi = lane >> 4;
  const int l15 = lane & 15;
  const int g = blockIdx.z;
  const float* Ag = A + (long)aG * g;
  const float* Bg = Bm + (long)bG * g;
  float* Cg = Cm + (long)cG * g;
  const int m0 = blockIdx.x * 128;
  const int n0 = blockIdx.y * 64;

  // async staging mapping (one b128 per issue)
  const int ar = tid >> 2;               // 0..63 (+64 on second issue)
  const int ac = (tid & 3) * 4;          // 0,4,8,12
  const int br = tid >> 4;               // 0..15
  const int bc = (tid & 15) * 4;         // 0..60

  auto issue_tile = [&](int k0, int buf) {
    async_copy_b128(lds_off(&As[buf][ar][ac]),      Ag + (m0 + ar) * lda + k0 + ac);
    async_copy_b128(lds_off(&As[buf][ar + 64][ac]), Ag + (m0 + ar + 64) * lda + k0 + ac);
    async_copy_b128(lds_off(&Bs[buf][br][bc]),      Bg + (k0 + br) * ldb + n0 + bc);
  };

  v8f acc[4];
#pragma unroll
  for (int t = 0; t < 4; ++t)
#pragma unroll
    for (int e = 0; e < 8; ++e) acc[t][e] = 0.f;

  const int nk = K >> 4;
  issue_tile(0, 0);
  wait_async0();
  __syncthreads();
  for (int kt = 0; kt < nk; ++kt) {
    const int buf = kt & 1;
    if (kt + 1 < nk) issue_tile((kt + 1) << 4, buf ^ 1);  // overlap DMA with WMMA
#pragma unroll
    for (int kk = 0; kk < 16; kk += 4) {
      v2f a;
      a.x = As[buf][wave * 16 + l15][kk + 2 * hi + 0];
      a.y = As[buf][wave * 16 + l15][kk + 2 * hi + 1];
#pragma unroll
      for (int t = 0; t < 4; ++t) {
        v2f b;
        b.x = Bs[buf][kk + 2 * hi + 0][t * 16 + l15];
        b.y = Bs[buf][kk + 2 * hi + 1][t * 16 + l15];
        acc[t] = WMMA_F32(a, b, acc[t]);
      }
    }
    wait_async0();
    __syncthreads();
  }
#pragma unroll
  for (int t = 0; t < 4; ++t) {
    int n = n0 + t * 16 + l15;
    float bv = bias ? bias[biasG * g + n] : 0.f;
#pragma unroll
    for (int v = 0; v < 8; ++v) {
      int m = m0 + wave * 16 + v + 8 * hi;
      Cg[m * ldc + n] = acc[t][v] + bv;
    }
  }
}

// ============================================================
// fused bilinear: per (c,dir) block-tile, T = maxout(h_bn[:,gsel,:]) @ Wcd,
// then per-row dot with h_bn[:,goth,:]; deterministic partials (no atomics).
// A tile is VALU-transformed (float4), B tile via async DMA.
__global__ __launch_bounds__(256) void wmma_bilinear_kernel(
    const float* __restrict__ hbn, const float* __restrict__ wt,
    const float* __restrict__ pw, const float* __restrict__ pb,
    float* __restrict__ bil) {
  __shared__ float As[128][20];
  __shared__ float Bs[16][68];
  const int tid = threadIdx.x;
  const int lane = tid & 31;
  const int wave = tid >> 5;
  const int hi = lane >> 4;
  const int l15 = lane & 15;
  const int cd = blockIdx.z;
  const int c = cd >> 1, dir = cd & 1;
  const int ga = c_aidx[c], gb = c_bidx[c];
  const int gsel = dir ? gb : ga;
  const int goth = dir ? ga : gb;
  const float pw0 = pw[c * 2 + 0], pw1 = pw[c * 2 + 1], pbc = pb[c];
  const int m0 = blockIdx.x * 128;
  const int n0 = blockIdx.y * 64;
  const float* W = wt + cd * (Hn * Hn);

  const int ar = tid >> 2;               // 0..63 (+64 second pass)
  const int ac = (tid & 3) * 4;
  const int br = tid >> 4;
  const int bc = (tid & 15) * 4;

  v8f acc[4];
#pragma unroll
  for (int t = 0; t < 4; ++t)
#pragma unroll
    for (int e = 0; e < 8; ++e) acc[t][e] = 0.f;

  for (int k0 = 0; k0 < Hn; k0 += 16) {
    async_copy_b128(lds_off(&Bs[br][bc]), W + (k0 + br) * Hn + n0 + bc);
#pragma unroll
    for (int p = 0; p < 2; ++p) {
      int r = ar + p * 64;
      float4 hv = *(const float4*)&hbn[((m0 + r) * Gn + gsel) * Hn + k0 + ac];
      float4 o;
      float d;
      d = hv.x - pbc; o.x = fmaxf(pw0 * d, pw1 * d);
      d = hv.y - pbc; o.y = fmaxf(pw0 * d, pw1 * d);
      d = hv.z - pbc; o.z = fmaxf(pw0 * d, pw1 * d);
      d = hv.w - pbc; o.w = fmaxf(pw0 * d, pw1 * d);
      *(float4*)&As[r][ac] = o;
    }
    wait_async0();
    __syncthreads();
#pragma unroll
    for (int kk = 0; kk < 16; kk += 4) {
      v2f a;
      a.x = As[wave * 16 + l15][kk + 2 * hi + 0];
      a.y = As[wave * 16 + l15][kk + 2 * hi + 1];
#pragma unroll
      for (int t = 0; t < 4; ++t) {
        v2f b;
        b.x = Bs[kk + 2 * hi + 0][t * 16 + l15];
        b.y = Bs[kk + 2 * hi + 1][t * 16 + l15];
        acc[t] = WMMA_F32(a, b, acc[t]);
      }
    }
    __syncthreads();
  }
  // dot epilogue: per output row b, s = sum_j T[b,j] * hbn[b,goth,j]
  const int slot = cd * 4 + blockIdx.y;   // 224 slots
#pragma unroll
  for (int v = 0; v < 8; ++v) {
    int b = m0 + wave * 16 + v + 8 * hi;
    float s = 0.f;
#pragma unroll
    for (int t = 0; t < 4; ++t) {
      int j = n0 + t * 16 + l15;
      s += acc[t][v] * hbn[(b * Gn + goth) * Hn + j];
    }
    s += __shfl_xor(s, 1, 16);
    s += __shfl_xor(s, 2, 16);
    s += __shfl_xor(s, 4, 16);
    s += __shfl_xor(s, 8, 16);
    if (l15 == 0) bil[slot * Bn + b] = s;
  }
}

// ============================================================
// big fused store: h_bn (output 2 + ws copy) and h_nl (output 3, 235 MB)
// one float4 (4 consecutive c) per thread -> b128 stores on the hot stream
__global__ __launch_bounds__(256) void hnl_kernel(
    const float* __restrict__ h, const float* __restrict__ g2,
    const float* __restrict__ b2, const float* __restrict__ mu2,
    const float* __restrict__ rstd2, const float* __restrict__ pw,
    const float* __restrict__ pb, float* __restrict__ out_hbn,
    float* __restrict__ hbn_ws, float* __restrict__ out_hnl) {
  unsigned t = blockIdx.x * 256u + threadIdx.x;  // < 8192*256*7
  unsigned n = t / 7u;
  unsigned k = t - n * 7u;
  int i = n & (Hn - 1);
  float hb = g2[i] * (h[n] - mu2[i]) * rstd2[i] + b2[i];
  int c0 = k * 4;
  float4 o;
  {
    float d = hb - pb[c0 + 0]; o.x = fmaxf(pw[2 * c0 + 0] * d, pw[2 * c0 + 1] * d);
  }
  {
    float d = hb - pb[c0 + 1]; o.y = fmaxf(pw[2 * c0 + 2] * d, pw[2 * c0 + 3] * d);
  }
  {
    float d = hb - pb[c0 + 2]; o.z = fmaxf(pw[2 * c0 + 4] * d, pw[2 * c0 + 5] * d);
  }
  {
    float d = hb - pb[c0 + 3]; o.w = fmaxf(pw[2 * c0 + 6] * d, pw[2 * c0 + 7] * d);
  }
  *(float4*)&out_hnl[(size_t)n * Cn + c0] = o;
  if (k == 0) { out_hbn[n] = hb; hbn_ws[n] = hb; }
}

// z maxout over adjacent pairs: (B,G,1024) -> (B,G,512), 4-wide
__global__ __launch_bounds__(256) void maxout_kernel(
    const float* __restrict__ z1, float* __restrict__ z1m) {
  int i4 = blockIdx.x * 256 + threadIdx.x;  // < 1024*8*512/4
  float4 a = *(const float4*)&z1[i4 * 8 + 0];
  float4 b = *(const float4*)&z1[i4 * 8 + 4];
  float4 o;
  o.x = fmaxf(a.x, a.y); o.y = fmaxf(a.z, a.w);
  o.z = fmaxf(b.x, b.y); o.w = fmaxf(b.z, b.w);
  *(float4*)&z1m[i4 * 4] = o;
}

// final: logits[b] = sum(bil partials) + sum_{g,o} zw0*hz^2 + zw1*hz + b0
__global__ __launch_bounds__(256) void final_kernel(
    const float* __restrict__ z2, const float* __restrict__ muz,
    const float* __restrict__ rstdz, const float* __restrict__ bnzg,
    const float* __restrict__ bnzb, const float* __restrict__ zw,
    const float* __restrict__ bil, const float* __restrict__ bsc,
    float* __restrict__ out) {
  __shared__ float red[256];
  const int b = blockIdx.x;
  const int t = threadIdx.x;
  float s = 0.f;
#pragma unroll
  for (int k = 0; k < 8; ++k) {
    int col = t + k * 256;  // 2048 cols = g*256+o
    float zv = z2[b * 2048 + col];
    float hz = bnzg[col] * (zv - muz[col]) * rstdz[col] + bnzb[col];
    s += zw[2 * col] * hz * hz + zw[2 * col + 1] * hz;
  }
  if (t < 224) s += bil[t * Bn + b];
  red[t] = s;
  __syncthreads();
  for (int off = 128; off > 0; off >>= 1) {
    if (t < off) red[t] += red[t + off];
    __syncthreads();
  }
  if (t == 0) out[b] = red[0] + bsc[0];
}

// ============================================================
extern "C" void kernel_launch(void* const* d_in, const int* in_sizes, int n_in,
                              void* d_out, int out_size, void* d_ws, size_t ws_size,
                              hipStream_t stream) {
  const float* x     = (const float*)d_in[0];
  const float* bn1_g = (const float*)d_in[1];
  const float* bn1_b = (const float*)d_in[2];
  const float* lin_w = (const float*)d_in[3];
  const float* lin_b = (const float*)d_in[4];
  const float* bn2_g = (const float*)d_in[5];
  const float* bn2_b = (const float*)d_in[6];
  const float* pw    = (const float*)d_in[7];
  const float* pb    = (const float*)d_in[8];
  const float* w     = (const float*)d_in[9];
  const float* zw    = (const float*)d_in[10];
  const float* wz1   = (const float*)d_in[11];
  const float* bz1   = (const float*)d_in[12];
  const float* wz2   = (const float*)d_in[13];
  const float* bnz_g = (const float*)d_in[14];
  const float* bnz_b = (const float*)d_in[15];
  const float* bsc   = (const float*)d_in[16];

  float* out = (float*)d_out;
  float* out_logits = out;                       // 1024
  float* out_hbn = out + Bn;                     // 2,097,152
  float* out_hnl = out + Bn + Nrows * Hn;        // 58,720,256

  float* ws = (float*)d_ws;
  float* mu1 = ws + WS_MU1;     float* rstd1 = ws + WS_RSTD1;
  float* mu2 = ws + WS_MU2;     float* rstd2 = ws + WS_RSTD2;
  float* muz = ws + WS_MUZ;     float* rstdz = ws + WS_RSTDZ;
  float* bil = ws + WS_BIL;
  float* Abuf = ws + WS_SHARED;  // reused later as z1
  float* z1   = ws + WS_SHARED;
  float* hbuf = ws + WS_H;
  float* hbn  = ws + WS_HBN;
  float* wt   = ws + WS_WT;
  float* z1m  = ws + WS_Z1M;
  float* z2   = ws + WS_Z2;

  // 1. BN1 stats over 8192 rows x 512 cols
  col_stats_kernel<<<HIDn / 32, 256, 0, stream>>>(x, Nrows, HIDn, mu1, rstd1);
  // 2. A = tanh(bn1(x))
  tanh_bn1_kernel<<<(Nrows * HIDn) / 1024, 256, 0, stream>>>(x, bn1_g, bn1_b, mu1, rstd1, Abuf);
  // 3. h = A @ lin_w + lin_b   (8192x512 @ 512x256)
  wmma_gemm_f32<<<dim3(Nrows / 128, Hn / 64, 1), 256, 0, stream>>>(
      Abuf, HIDn, 0, lin_w, Hn, 0, hbuf, Hn, 0, lin_b, 0, HIDn);
  // 4. BN2 stats
  col_stats_kernel<<<Hn / 32, 256, 0, stream>>>(hbuf, Nrows, Hn, mu2, rstd2);
  // 5. repack w -> wt[(c*2+dir), i, j]
  repack_w_kernel<<<(2 * Cn * Hn * Hn) / 256, 256, 0, stream>>>(w, wt);
  // 6. h_bn + h_nl (big store, b128)
  hnl_kernel<<<(Nrows * Hn * (Cn / 4)) / 256, 256, 0, stream>>>(
      hbuf, bn2_g, bn2_b, mu2, rstd2, pw, pb, out_hbn, hbn, out_hnl);
  // 7. bilinear logits partials
  wmma_bilinear_kernel<<<dim3(Bn / 128, Hn / 64, 2 * Cn), 256, 0, stream>>>(
      hbn, wt, pw, pb, bil);
  // 8. z1 = h_bn @ wz1 + bz1 per group  (1024x256 @ 256x1024, g=8)
  wmma_gemm_f32<<<dim3(Bn / 128, Z1n / 64, Gn), 256, 0, stream>>>(
      hbn, Gn * Hn, Hn, wz1, Z1n, Hn * Z1n, z1, Gn * Z1n, Z1n, bz1, Z1n, Hn);
  // 9. pairwise maxout -> z1m (B,G,512)
  maxout_kernel<<<(Bn * Gn * Z2n) / 1024, 256, 0, stream>>>(z1, z1m);
  // 10. z2 = z1m @ wz2 per group  (1024x512 @ 512x256, g=8)
  wmma_gemm_f32<<<dim3(Bn / 128, Hn / 64, Gn), 256, 0, stream>>>(
      z1m, Gn * Z2n, Z2n, wz2, Hn, Z2n * Hn, z2, Gn * Hn, Hn, nullptr, 0, Z2n);
  // 11. BNz stats: z2 viewed as (1024 x 2048)
  col_stats_kernel<<<(Gn * Hn) / 32, 256, 0, stream>>>(z2, Bn, Gn * Hn, muz, rstdz);
  // 12. final logits
  final_kernel<<<Bn, 256, 0, stream>>>(z2, muz, rstdz, bnz_g, bnz_b, zw, bil, bsc, out_logits);
}